// PreprocessorGPU_80135499808884
// MI455X (gfx1250) — compile-verified
//
#include <hip/hip_runtime.h>
#include <hip/hip_bf16.h>

// ---------------- static config (mirrors the reference) ----------------
#define BB       2
#define NN       8192
#define KK       40
#define LL       3
#define SENT_F   8192.0f
#define GRIDC    17                  // ceil(1.0/0.06) cells per axis (covers both dl's)
#define NC       (GRIDC*GRIDC*GRIDC) // 4913 voxel cells

typedef __attribute__((ext_vector_type(2))) float v2f;
typedef __attribute__((ext_vector_type(8))) float v8f;

// ---------------- generic fills / copies ----------------
__global__ void fill_f32_kernel(float* __restrict__ dst, float v, int n) {
  int i = blockIdx.x * blockDim.x + threadIdx.x;
  if (i < n) dst[i] = v;
}
__global__ void fill_i32_kernel(int* __restrict__ dst, int v, int n) {
  int i = blockIdx.x * blockDim.x + threadIdx.x;
  if (i < n) dst[i] = v;
}
__global__ void copy_f32_kernel(const float* __restrict__ src, float* __restrict__ dst, int n) {
  int i = blockIdx.x * blockDim.x + threadIdx.x;
  if (i < n) dst[i] = src[i];
}

// ---------------- voxel-grid subsample ----------------
// Deterministic dense-grid reduction: one thread owns one voxel cell and scans
// all points in original order (matches the reference's stable segment sums).
__global__ void voxel_accum_kernel(const float* __restrict__ pts,
                                   const int* __restrict__ pmask,
                                   float* __restrict__ cellsum,  // [B][NC][4] = sx,sy,sz,cnt
                                   float dl) {
  int c = blockIdx.x * blockDim.x + threadIdx.x;
  int b = blockIdx.y;
  if (c >= NC) return;
  int cz = c % GRIDC;
  int cy = (c / GRIDC) % GRIDC;
  int cx = c / (GRIDC * GRIDC);
  float sx = 0.f, sy = 0.f, sz = 0.f, sc = 0.f;
  for (int i = 0; i < NN; ++i) {
    if (pmask && pmask[b * NN + i] == 0) continue;
    const float* p = pts + ((size_t)b * NN + i) * 3;
    float px = p[0], py = p[1], pz = p[2];
    int vx = (int)floorf(px / dl);
    int vy = (int)floorf(py / dl);
    int vz = (int)floorf(pz / dl);
    if (vx == cx && vy == cy && vz == cz) { sx += px; sy += py; sz += pz; sc += 1.f; }
  }
  float* cs = cellsum + ((size_t)b * NC + c) * 4;
  cs[0] = sx; cs[1] = sy; cs[2] = sz; cs[3] = sc;
}

// Serial rank/emit: ascending compact cell index == ascending reference voxel
// key (both lexicographic in (cx,cy,cz), cells < 4096). One thread per batch.
__global__ void voxel_emit_kernel(const float* __restrict__ cellsum,
                                  float* __restrict__ pool_p,  // [B][N][3], pre-zeroed
                                  int* __restrict__ pool_m,    // [B][N],   pre-zeroed
                                  int* __restrict__ plen) {    // [B]
  int b = blockIdx.x;
  if (threadIdx.x != 0) return;
  int acc = 0;
  for (int c = 0; c < NC; ++c) {
    const float* cs = cellsum + ((size_t)b * NC + c) * 4;
    float cnt = cs[3];
    if (cnt > 0.f) {
      float inv = 1.0f / cnt;
      float* o = pool_p + ((size_t)b * NN + acc) * 3;
      o[0] = cs[0] * inv; o[1] = cs[1] * inv; o[2] = cs[2] * inv;
      pool_m[b * NN + acc] = 1;
      ++acc;
    }
  }
  plen[b] = acc;
}

// ---------------- WMMA ball query ----------------
// One wave (32 lanes) per 32 queries. d^2 = |q|^2 + |s|^2 - 2 q.s with the
// q.s term computed by V_WMMA_F32_16X16X4_F32 (two per 16-support tile).
__global__ __launch_bounds__(32)
void ballq_wmma_kernel(const float* __restrict__ qpts,
                       const float* __restrict__ spts,
                       const int* __restrict__ qmask,
                       const int* __restrict__ smask,
                       float* __restrict__ out,   // [B][N][K] as float indices
                       float r2) {
  __shared__ float qxL[32], qyL[32], qzL[32], qn[32];
  __shared__ float dtile[32 * 16];
  __shared__ int   smtile[16];
  __shared__ float bd[32 * KK];
  __shared__ int   bi[32 * KK];

  const int lane = threadIdx.x;
  const int b    = blockIdx.y;
  const int q0   = blockIdx.x * 32;

  // Load this wave's 32 query points.
  {
    const float* qp = qpts + ((size_t)b * NN + q0 + lane) * 3;
    float x = qp[0], y = qp[1], z = qp[2];
    qxL[lane] = x; qyL[lane] = y; qzL[lane] = z;
    qn[lane]  = x * x + y * y + z * z;
  }
  __syncthreads();

  // A matrices per ISA 16x4 f32 layout: lanes 0-15 -> (K0=x, K1=y),
  // lanes 16-31 -> (K2=z, K3=0).
  const int mq = lane & 15;
  v2f alo, ahi;
  if (lane < 16) {
    alo.x = qxL[mq];      alo.y = qyL[mq];
    ahi.x = qxL[16 + mq]; ahi.y = qyL[16 + mq];
  } else {
    alo.x = qzL[mq];      alo.y = 0.f;
    ahi.x = qzL[16 + mq]; ahi.y = 0.f;
  }

  int   cnt = 0;
  float maxv = -1.f;
  int   maxp = 0;
  const int base = lane * KK;

  for (int s0 = 0; s0 < NN; s0 += 16) {
    if (s0 + 16 < NN)
      __builtin_prefetch(spts + ((size_t)b * NN + s0 + 16) * 3, 0, 0);

    // Each lane loads support (s0 + lane&15); builds B per 4x16 f32 layout.
    const int sidx = s0 + (lane & 15);
    const float* sp = spts + ((size_t)b * NN + sidx) * 3;
    float sx = sp[0], sy = sp[1], sz = sp[2];
    float sn2 = sx * sx + sy * sy + sz * sz;
    int sm_ok = smask ? smask[b * NN + sidx] : 1;
    if (lane < 16) smtile[lane] = sm_ok;

    v2f bmat;
    bmat.x = (lane < 16) ? sx : sz;
    bmat.y = (lane < 16) ? sy : 0.f;

    v8f clo = {};
    v8f chi = {};
    clo = __builtin_amdgcn_wmma_f32_16x16x4_f32(false, alo, false, bmat,
                                                (short)0, clo, false, false);
    chi = __builtin_amdgcn_wmma_f32_16x16x4_f32(false, ahi, false, bmat,
                                                (short)0, chi, false, false);

    // Unpack C layout: element j on this lane is (row = j + 8*(lane>=16),
    // col = lane&15); hi-WMMA rows are +16.
    const int n = lane & 15;
    const int mbase = (lane >> 4) * 8;
#pragma unroll
    for (int j = 0; j < 8; ++j) {
      float dlo = qn[mbase + j]      + sn2 - 2.f * clo[j];
      float dhi = qn[16 + mbase + j] + sn2 - 2.f * chi[j];
      dtile[(mbase + j) * 16 + n]      = dlo;
      dtile[(16 + mbase + j) * 16 + n] = dhi;
    }
    __syncthreads();

    // Selection: lane t owns query row t; maintain radius-filtered top-K.
    for (int n2 = 0; n2 < 16; ++n2) {
      if (!smtile[n2]) continue;
      float d2v = dtile[lane * 16 + n2];
      if (d2v <= r2) {
        int idx = s0 + n2;
        if (cnt < KK) {
          bd[base + cnt] = d2v; bi[base + cnt] = idx;
          if (cnt == 0 || d2v > maxv) { maxv = d2v; maxp = cnt; }
          ++cnt;
        } else if (d2v < maxv) {
          bd[base + maxp] = d2v; bi[base + maxp] = idx;
          maxv = -1.f;
          for (int k = 0; k < KK; ++k) {
            float dv = bd[base + k];
            if (dv > maxv) { maxv = dv; maxp = k; }
          }
        }
      }
    }
    __syncthreads();
  }

  // Insertion sort ascending by (dist, idx) — matches top_k(-d2) ordering.
  for (int i = 1; i < cnt; ++i) {
    float dv = bd[base + i];
    int   iv = bi[base + i];
    int j = i - 1;
    while (j >= 0 && (bd[base + j] > dv ||
                      (bd[base + j] == dv && bi[base + j] > iv))) {
      bd[base + j + 1] = bd[base + j];
      bi[base + j + 1] = bi[base + j];
      --j;
    }
    bd[base + j + 1] = dv;
    bi[base + j + 1] = iv;
  }

  const int qgi = q0 + lane;
  const bool qok = qmask ? (qmask[b * NN + qgi] != 0) : true;
  float* dst = out + ((size_t)b * NN + qgi) * KK;
  for (int k = 0; k < KK; ++k)
    dst[k] = (qok && k < cnt) ? (float)bi[base + k] : SENT_F;
}

// ---------------- lengths ----------------
__global__ void len_write_kernel(const int* __restrict__ len0,
                                 const int* __restrict__ len1,
                                 float* __restrict__ out_len) {
  int b = threadIdx.x;
  if (b >= BB) return;
  out_len[0 * BB + b] = (float)NN;
  out_len[1 * BB + b] = (float)len0[b];
  out_len[2 * BB + b] = (float)len1[b];
}

// ---------------- host orchestration ----------------
extern "C" void kernel_launch(void* const* d_in, const int* in_sizes, int n_in,
                              void* d_out, int out_size, void* d_ws, size_t ws_size,
                              hipStream_t stream) {
  (void)in_sizes; (void)n_in; (void)out_size; (void)ws_size;
  const float* pts = (const float*)d_in[0];
  float* outf = (float*)d_out;

  const int PSZ  = LL * BB * NN * 3;       // points slab
  const int ISL  = BB * NN * KK;           // one layer of one index tensor
  float* out_points = outf;
  float* out_neigh  = outf + PSZ;
  float* out_pool   = out_neigh + LL * ISL;
  float* out_up     = out_pool  + LL * ISL;
  float* out_len    = out_up    + LL * ISL;

  // workspace carve
  char* w = (char*)d_ws;
  float* pool_p0 = (float*)w; w += (size_t)BB * NN * 3 * sizeof(float);
  int*   pool_m0 = (int*)w;   w += (size_t)BB * NN * sizeof(int);
  float* pool_p1 = (float*)w; w += (size_t)BB * NN * 3 * sizeof(float);
  int*   pool_m1 = (int*)w;   w += (size_t)BB * NN * sizeof(int);
  int*   len0    = (int*)w;   w += (size_t)BB * sizeof(int);
  int*   len1    = (int*)w;   w += (size_t)BB * sizeof(int);
  float* cellsum = (float*)w; w += (size_t)BB * NC * 4 * sizeof(float);

  const float R0 = 0.075f, R1 = 0.15f, R2 = 0.30f;
  const float DL0 = 0.06f, DL1 = 0.12f;

  const int n3 = BB * NN * 3, n1 = BB * NN;
  dim3 bq_grid(NN / 32, BB);
  dim3 vox_grid((NC + 127) / 128, BB);

  // zero pool buffers (emit only writes valid rows)
  fill_f32_kernel<<<(n3 + 255) / 256, 256, 0, stream>>>(pool_p0, 0.f, n3);
  fill_i32_kernel<<<(n1 + 255) / 256, 256, 0, stream>>>(pool_m0, 0, n1);
  fill_f32_kernel<<<(n3 + 255) / 256, 256, 0, stream>>>(pool_p1, 0.f, n3);
  fill_i32_kernel<<<(n1 + 255) / 256, 256, 0, stream>>>(pool_m1, 0, n1);

  // ---- layer 0 (points = pts, all valid) ----
  ballq_wmma_kernel<<<bq_grid, 32, 0, stream>>>(pts, pts, nullptr, nullptr,
                                                out_neigh + 0 * ISL, R0 * R0);
  voxel_accum_kernel<<<vox_grid, 128, 0, stream>>>(pts, nullptr, cellsum, DL0);
  voxel_emit_kernel<<<BB, 32, 0, stream>>>(cellsum, pool_p0, pool_m0, len0);
  ballq_wmma_kernel<<<bq_grid, 32, 0, stream>>>(pool_p0, pts, pool_m0, nullptr,
                                                out_pool + 0 * ISL, R0 * R0);
  ballq_wmma_kernel<<<bq_grid, 32, 0, stream>>>(pts, pool_p0, nullptr, pool_m0,
                                                out_up + 0 * ISL, (2.f * R0) * (2.f * R0));

  // ---- layer 1 (points = pool_p0) ----
  ballq_wmma_kernel<<<bq_grid, 32, 0, stream>>>(pool_p0, pool_p0, pool_m0, pool_m0,
                                                out_neigh + 1 * ISL, R1 * R1);
  voxel_accum_kernel<<<vox_grid, 128, 0, stream>>>(pool_p0, pool_m0, cellsum, DL1);
  voxel_emit_kernel<<<BB, 32, 0, stream>>>(cellsum, pool_p1, pool_m1, len1);
  ballq_wmma_kernel<<<bq_grid, 32, 0, stream>>>(pool_p1, pool_p0, pool_m1, pool_m0,
                                                out_pool + 1 * ISL, R1 * R1);
  ballq_wmma_kernel<<<bq_grid, 32, 0, stream>>>(pool_p0, pool_p1, pool_m0, pool_m1,
                                                out_up + 1 * ISL, (2.f * R1) * (2.f * R1));

  // ---- layer 2 (points = pool_p1, no stride -> sentinel pool/up) ----
  ballq_wmma_kernel<<<bq_grid, 32, 0, stream>>>(pool_p1, pool_p1, pool_m1, pool_m1,
                                                out_neigh + 2 * ISL, R2 * R2);
  fill_f32_kernel<<<(ISL + 255) / 256, 256, 0, stream>>>(out_pool + 2 * ISL, SENT_F, ISL);
  fill_f32_kernel<<<(ISL + 255) / 256, 256, 0, stream>>>(out_up + 2 * ISL, SENT_F, ISL);

  // ---- per-layer points ----
  copy_f32_kernel<<<(n3 + 255) / 256, 256, 0, stream>>>(pts,     out_points + 0 * n3, n3);
  copy_f32_kernel<<<(n3 + 255) / 256, 256, 0, stream>>>(pool_p0, out_points + 1 * n3, n3);
  copy_f32_kernel<<<(n3 + 255) / 256, 256, 0, stream>>>(pool_p1, out_points + 2 * n3, n3);

  // ---- lengths ----
  len_write_kernel<<<1, 32, 0, stream>>>(len0, len1, out_len);
}